// AdaptiveGraphSAGE_65996467470992
// MI455X (gfx1250) — compile-verified
//
#include <hip/hip_runtime.h>
#include <math.h>

#define FEAT 128
#define EPSBN 1e-5f

typedef __attribute__((ext_vector_type(2))) float v2f;
typedef __attribute__((ext_vector_type(8))) float v8f;

// ---- native CDNA5 f32 vector-memory atomics (no CAS loops) ----
__device__ __forceinline__ void gatomic_add_f32(float* p, float v) {
  asm volatile("global_atomic_add_f32 %0, %1, off" : : "v"(p), "v"(v) : "memory");
}
__device__ __forceinline__ void gatomic_max_f32(float* p, float v) {
  asm volatile("global_atomic_max_num_f32 %0, %1, off" : : "v"(p), "v"(v) : "memory");
}

// ---- init: sum=0, max=-inf, optionally cnt=0 / emb=0 ----
__global__ __launch_bounds__(256)
void k_init(float* __restrict__ sum, float* __restrict__ mx,
            float* cnt, float* emb, int nf, int n) {
  int i = blockIdx.x * 256 + threadIdx.x;
  if (i < nf) { sum[i] = 0.f; mx[i] = -INFINITY; }
  if (cnt && i < n) cnt[i] = 0.f;
  if (emb && i < FEAT) emb[i] = 0.f;
}

// ---- in-degree count ----
__global__ __launch_bounds__(256)
void k_count(const int* __restrict__ dst, float* __restrict__ cnt, int e) {
  int i = blockIdx.x * 256 + threadIdx.x;
  if (i < e) gatomic_add_f32(cnt + dst[i], 1.f);
}

// ---- edge scatter: one wave per edge; lane owns 4 contiguous feats ----
__global__ __launch_bounds__(256)
void k_scatter(const float* __restrict__ H, const int* __restrict__ src,
               const int* __restrict__ dst, float* __restrict__ Sum,
               float* __restrict__ Max, int e) {
  int edge = blockIdx.x * 8 + (threadIdx.x >> 5);
  if (edge >= e) return;
  int lane = threadIdx.x & 31;
  int s = src[edge], d = dst[edge];
  const float4 v = *(const float4*)(H + (size_t)s * FEAT + lane * 4);
  float* ps = Sum + (size_t)d * FEAT + lane * 4;
  float* pm = Max + (size_t)d * FEAT + lane * 4;
  gatomic_add_f32(ps + 0, v.x); gatomic_add_f32(ps + 1, v.y);
  gatomic_add_f32(ps + 2, v.z); gatomic_add_f32(ps + 3, v.w);
  gatomic_max_f32(pm + 0, v.x); gatomic_max_f32(pm + 1, v.y);
  gatomic_max_f32(pm + 2, v.z); gatomic_max_f32(pm + 3, v.w);
}

// ---- finalize aggregates: mean = sum/max(cnt,1); max: -inf -> 0 ----
__global__ __launch_bounds__(256)
void k_finalize(float* __restrict__ sum, float* __restrict__ mx,
                const float* __restrict__ cnt, int nf) {
  int i = blockIdx.x * 256 + threadIdx.x;
  if (i >= nf) return;
  float c = cnt[i >> 7];
  sum[i] = sum[i] / fmaxf(c, 1.f);
  float m = mx[i];
  mx[i] = (m > -3.0e38f) ? m : 0.f;
}

// ---- fused fp32 WMMA GEMM: Out = Agg@Wl + H@Wr + bias ----
// One wave computes one 16x16 output tile via V_WMMA_F32_16X16X4_F32.
// A frag (16x4 f32): lane 0-15 row M=lane holds K=0,1; lane 16-31 holds K=2,3.
// B frag (4x16 f32): mirrored (lane%16 = col, lane/16 selects K pair).
// C/D (16x16 f32): 8 VGPRs, vgpr r -> row r + 8*(lane/16), col = lane%16.
__global__ __launch_bounds__(256)
void k_gemm(const float* __restrict__ Agg, const float* __restrict__ H,
            const float* __restrict__ Wl, const float* __restrict__ Wr,
            const float* __restrict__ bias, float* __restrict__ Out, int nrows) {
  const int wave = threadIdx.x >> 5;   // 0..7 -> N tile
  const int lane = threadIdx.x & 31;
  const int m0 = blockIdx.x * 16;
  const int n0 = wave * 16;
  if (m0 >= nrows) return;
  const int row   = m0 + (lane & 15);
  const int col   = n0 + (lane & 15);
  const int khalf = (lane >> 4) * 2;   // 0 or 2

  v8f acc = {};
  #pragma unroll 4
  for (int k = 0; k < FEAT; k += 4) {
    v2f a = *(const v2f*)(Agg + (size_t)row * FEAT + k + khalf);
    v2f b;
    b.x = Wl[(size_t)(k + khalf) * FEAT + col];
    b.y = Wl[(size_t)(k + khalf + 1) * FEAT + col];
    acc = __builtin_amdgcn_wmma_f32_16x16x4_f32(false, a, false, b,
                                                (short)0, acc, false, false);
  }
  #pragma unroll 4
  for (int k = 0; k < FEAT; k += 4) {
    v2f a = *(const v2f*)(H + (size_t)row * FEAT + k + khalf);
    v2f b;
    b.x = Wr[(size_t)(k + khalf) * FEAT + col];
    b.y = Wr[(size_t)(k + khalf + 1) * FEAT + col];
    acc = __builtin_amdgcn_wmma_f32_16x16x4_f32(false, a, false, b,
                                                (short)0, acc, false, false);
  }
  const int rbase = m0 + ((lane >> 4) << 3);
  const float bia = bias[col];
  #pragma unroll
  for (int r = 0; r < 8; ++r)
    Out[(size_t)(rbase + r) * FEAT + col] = acc[r] + bia;
}

// ---- per-node epilogue: L2-normalize, BN(eval), ReLU per branch; combine ----
__global__ __launch_bounds__(256)
void k_epilogue(const float* __restrict__ Om, const float* __restrict__ Ox,
                const float* gm, const float* bm, const float* mm, const float* vm_,
                const float* gx, const float* bx, const float* mxp, const float* vx_,
                const float* aw, float* __restrict__ Hout, int n) {
  int node = blockIdx.x * 8 + (threadIdx.x >> 5);
  if (node >= n) return;
  int lane = threadIdx.x & 31;
  float w0 = aw[0], w1 = aw[1];
  float wm = fmaxf(w0, w1);
  float e0 = expf(w0 - wm), e1 = expf(w1 - wm);
  float sinv = 1.f / (e0 + e1);
  w0 = e0 * sinv; w1 = e1 * sinv;

  float am[4], ax[4], ssm = 0.f, ssx = 0.f;
  #pragma unroll
  for (int r = 0; r < 4; ++r) {
    int f = lane + 32 * r;
    am[r] = Om[(size_t)node * FEAT + f]; ssm += am[r] * am[r];
    ax[r] = Ox[(size_t)node * FEAT + f]; ssx += ax[r] * ax[r];
  }
  #pragma unroll
  for (int off = 16; off > 0; off >>= 1) {   // wave32 reduction
    ssm += __shfl_xor(ssm, off, 32);
    ssx += __shfl_xor(ssx, off, 32);
  }
  float im = 1.f / fmaxf(sqrtf(ssm), 1e-12f);
  float ix = 1.f / fmaxf(sqrtf(ssx), 1e-12f);
  #pragma unroll
  for (int r = 0; r < 4; ++r) {
    int f = lane + 32 * r;
    float a = am[r] * im;
    a = (a - mm[f]) * rsqrtf(vm_[f] + EPSBN) * gm[f] + bm[f];
    a = fmaxf(a, 0.f);
    float c = ax[r] * ix;
    c = (c - mxp[f]) * rsqrtf(vx_[f] + EPSBN) * gx[f] + bx[f];
    c = fmaxf(c, 0.f);
    Hout[(size_t)node * FEAT + f] = w0 * a + w1 * c;
  }
}

// ---- column-wise partial sums for mean pooling ----
__global__ __launch_bounds__(128)
void k_reduce_mean(const float* __restrict__ H, float* __restrict__ emb, int n) {
  int d = threadIdx.x;                 // 0..127
  int start = blockIdx.x * 512;
  int end = min(start + 512, n);
  float s = 0.f;
  for (int i = start; i < end; ++i) s += H[(size_t)i * FEAT + d];
  gatomic_add_f32(emb + d, s);
}

// ---- tiny MLP head: 128 -> 256 -> 128 -> 2, single block ----
__global__ __launch_bounds__(256)
void k_classifier(const float* __restrict__ emb,
                  const float* W1, const float* b1, const float* g1,
                  const float* be1, const float* m1, const float* v1,
                  const float* W2, const float* b2, const float* g2,
                  const float* be2, const float* m2, const float* v2,
                  const float* W3, const float* b3,
                  float* __restrict__ out, float invN) {
  __shared__ float z0[128], z1[256], z2[128];
  int t = threadIdx.x;
  if (t < 128) { float e = emb[t] * invN; z0[t] = e; out[2 + t] = e; }
  __syncthreads();
  {
    float s = b1[t];
    for (int k = 0; k < 128; ++k) s += z0[k] * W1[k * 256 + t];
    s = (s - m1[t]) * rsqrtf(v1[t] + EPSBN) * g1[t] + be1[t];
    z1[t] = fmaxf(s, 0.f);
  }
  __syncthreads();
  if (t < 128) {
    float s = b2[t];
    for (int k = 0; k < 256; ++k) s += z1[k] * W2[k * 128 + t];
    s = (s - m2[t]) * rsqrtf(v2[t] + EPSBN) * g2[t] + be2[t];
    z2[t] = fmaxf(s, 0.f);
  }
  __syncthreads();
  if (t < 2) {
    float s = b3[t];
    for (int k = 0; k < 128; ++k) s += z2[k] * W3[k * 2 + t];
    out[t] = s;
  }
}

#define IN(i) ((const float*)d_in[(i)])

extern "C" void kernel_launch(void* const* d_in, const int* in_sizes, int n_in,
                              void* d_out, int out_size, void* d_ws, size_t ws_size,
                              hipStream_t stream) {
  const float* x = (const float*)d_in[0];
  const int* ei  = (const int*)d_in[1];
  const int N = in_sizes[0] / FEAT;
  const int E = in_sizes[1] / 2;
  const int* src = ei;
  const int* dst = ei + E;

  // workspace layout (floats)
  const size_t NF = (size_t)N * FEAT;
  float* ws      = (float*)d_ws;
  float* buf_h   = ws;                 // [N,128] current hidden
  float* buf_sum = ws + NF;            // [N,128] mean agg (in-place)
  float* buf_max = ws + 2 * NF;        // [N,128] max agg (in-place)
  float* buf_om  = ws + 3 * NF;        // [N,128] mean-branch GEMM out
  float* buf_ox  = ws + 4 * NF;        // [N,128] max-branch GEMM out
  float* cnt     = ws + 5 * NF;        // [N]
  float* emb     = ws + 5 * NF + N;    // [128]

  const int gridNF = (int)((NF + 255) / 256);
  // flattened param bases (setup_inputs insertion order):
  // layer l: mean{Wl,bl,Wr,gamma,beta,rmean,rvar}, max{...}, aggr_w
  const int mb[2] = {2, 17}, xb[2] = {9, 24}, wb[2] = {16, 31};

  for (int l = 0; l < 2; ++l) {
    const float* hin = (l == 0) ? x : buf_h;
    k_init<<<gridNF, 256, 0, stream>>>(buf_sum, buf_max,
                                       l == 0 ? cnt : nullptr,
                                       l == 0 ? emb : nullptr, (int)NF, N);
    if (l == 0)
      k_count<<<(E + 255) / 256, 256, 0, stream>>>(dst, cnt, E);
    k_scatter<<<(E + 7) / 8, 256, 0, stream>>>(hin, src, dst, buf_sum, buf_max, E);
    k_finalize<<<gridNF, 256, 0, stream>>>(buf_sum, buf_max, cnt, (int)NF);
    // mean branch: agg_mean@Wl + h@Wr + bl
    k_gemm<<<N / 16, 256, 0, stream>>>(buf_sum, hin, IN(mb[l] + 0), IN(mb[l] + 2),
                                       IN(mb[l] + 1), buf_om, N);
    // max branch
    k_gemm<<<N / 16, 256, 0, stream>>>(buf_max, hin, IN(xb[l] + 0), IN(xb[l] + 2),
                                       IN(xb[l] + 1), buf_ox, N);
    k_epilogue<<<(N + 7) / 8, 256, 0, stream>>>(
        buf_om, buf_ox,
        IN(mb[l] + 3), IN(mb[l] + 4), IN(mb[l] + 5), IN(mb[l] + 6),
        IN(xb[l] + 3), IN(xb[l] + 4), IN(xb[l] + 5), IN(xb[l] + 6),
        IN(wb[l]), buf_h, N);
  }

  k_reduce_mean<<<(N + 511) / 512, 128, 0, stream>>>(buf_h, emb, N);
  k_classifier<<<1, 256, 0, stream>>>(
      emb,
      IN(32), IN(33), IN(34), IN(35), IN(36), IN(37),
      IN(38), IN(39), IN(40), IN(41), IN(42), IN(43),
      IN(44), IN(45),
      (float*)d_out, 1.0f / (float)N);
}